// QuanvolutionHybrid_65481071407105
// MI455X (gfx1250) — compile-verified
//
#include <hip/hip_runtime.h>
#include <math.h>

// Quanvolution hybrid for MI455X (gfx1250, wave32).
// Stage 1: fold both weight layers (RY/RZ/CNOT x2) into one 16x16 complex
//          unitary V (weights are shared across patches).
// Stage 2: per 16-patch tile, Psi (16 amps x 16 patches, real) ->
//          Ur = Vr*Psi, Ui = Vi*Psi via V_WMMA_F32_16X16X4_F32 (fp32 exact),
//          probs = Ur^2+Ui^2, signed Z-sums -> qf (2048 x 784).
// Stage 3: logits = qf @ Wc^T + bc via the same f32 WMMA (K=784 in 196 steps),
//          then log_softmax.

typedef __attribute__((ext_vector_type(2))) float v2f;
typedef __attribute__((ext_vector_type(8))) float v8f;

#define NPATCH (2048 * 196)   // 401408
#define NTILE  (NPATCH / 16)  // 25088
#define QWAVES 8              // waves per block in quanvolve
#define CWAVES 4              // waves per block in classifier

// ---------------------------------------------------------------------------
// Kernel 0: build the 16x16 layer unitary V from weights (L=2, shape (2,4,2)).
// Thread j computes column V|j>. Wire q <-> bit (3-q) of the flat index.
// ---------------------------------------------------------------------------
__global__ void build_unitary(const float* __restrict__ w,
                              float* __restrict__ Vr, float* __restrict__ Vi) {
  int j = threadIdx.x;
  float ar[16], ai[16];
  for (int i = 0; i < 16; ++i) { ar[i] = 0.f; ai[i] = 0.f; }
  ar[j] = 1.f;

  for (int l = 0; l < 2; ++l) {
    for (int q = 0; q < 4; ++q) {
      int mask = 1 << (3 - q);
      // RY(theta): [[c,-s],[s,c]], c=cos(th/2), s=sin(th/2)
      float th = w[(l * 4 + q) * 2 + 0];
      float c = cosf(0.5f * th), s = sinf(0.5f * th);
      for (int i = 0; i < 16; ++i) {
        if (!(i & mask)) {
          int i1 = i | mask;
          float r0 = ar[i], m0 = ai[i], r1 = ar[i1], m1 = ai[i1];
          ar[i]  = c * r0 - s * r1;  ai[i]  = c * m0 - s * m1;
          ar[i1] = s * r0 + c * r1;  ai[i1] = s * m0 + c * m1;
        }
      }
      // RZ(phi): diag(e^{-i phi/2}, e^{+i phi/2})
      float ph = w[(l * 4 + q) * 2 + 1];
      float er = cosf(0.5f * ph), ei = -sinf(0.5f * ph);
      for (int i = 0; i < 16; ++i) {
        float r = ar[i], m = ai[i];
        if (i & mask) {  // multiply by conj(e) = (er, -ei)
          ar[i] = r * er + m * ei;
          ai[i] = m * er - r * ei;
        } else {         // multiply by e = (er, ei)
          ar[i] = r * er - m * ei;
          ai[i] = m * er + r * ei;
        }
      }
    }
    // CNOT(0,1): control bit3, target bit2 -> swap target where control=1
    for (int i = 0; i < 16; ++i) {
      if ((i & 8) && !(i & 4)) {
        int i1 = i | 4;
        float tr = ar[i], ti = ai[i];
        ar[i] = ar[i1]; ai[i] = ai[i1];
        ar[i1] = tr;    ai[i1] = ti;
      }
    }
    // CNOT(2,3): control bit1, target bit0
    for (int i = 0; i < 16; ++i) {
      if ((i & 2) && !(i & 1)) {
        int i1 = i | 1;
        float tr = ar[i], ti = ai[i];
        ar[i] = ar[i1]; ai[i] = ai[i1];
        ar[i1] = tr;    ai[i1] = ti;
      }
    }
  }
  for (int i = 0; i < 16; ++i) {
    Vr[i * 16 + j] = ar[i];
    Vi[i * 16 + j] = ai[i];
  }
}

// product-state amplitude: bit3<->wire0 ... bit0<->wire3
__device__ __forceinline__ float psi_at(int k, float c0, float s0, float c1,
                                        float s1, float c2, float s2, float c3,
                                        float s3) {
  float v = (k & 8) ? s0 : c0;
  v *= (k & 4) ? s1 : c1;
  v *= (k & 2) ? s2 : c2;
  v *= (k & 1) ? s3 : c3;
  return v;
}

// ---------------------------------------------------------------------------
// Kernel 1: quanvolution. One wave handles 16 patches (one WMMA tile).
// 32-bit WMMA operand layouts (ISA 7.12.2):
//   A 16x4 : lane -> M = lane&15, VGPR r -> K = r + 2*(lane>>4)
//   B 4x16 : lane -> N = lane&15, VGPR r -> K = r + 2*(lane>>4)  (mirrored)
//   C 16x16: lane -> N = lane&15, VGPR r -> M = r + 8*(lane>>4)
// ---------------------------------------------------------------------------
__global__ __launch_bounds__(QWAVES * 32) void quanvolve(
    const float* __restrict__ x, const float* __restrict__ Vg,
    float* __restrict__ qf) {
  __shared__ float sV[512];  // Vr[0..255], Vi[256..511]
  for (int t = threadIdx.x; t < 512; t += blockDim.x) sV[t] = Vg[t];
  __syncthreads();
  const float* sVr = sV;
  const float* sVi = sV + 256;

  int wave = threadIdx.x >> 5;
  int lane = threadIdx.x & 31;
  int h = lane >> 4;      // half-wave select
  int col = lane & 15;    // A-row M  and  B-column (patch within tile)

  int tile = blockIdx.x * QWAVES + wave;
  int patch = tile * 16 + col;
  int b = patch / 196;
  int r196 = patch - b * 196;
  int py = r196 / 14;
  int px = r196 - py * 14;

  // 2x2 patch, row-major: d0=(0,0) d1=(0,1) d2=(1,0) d3=(1,1)
  const float* xp = x + b * 784 + (py * 2) * 28 + px * 2;
  float d0 = xp[0], d1 = xp[1], d2 = xp[28], d3 = xp[29];
  float c0 = cosf(0.5f * d0), s0 = sinf(0.5f * d0);
  float c1 = cosf(0.5f * d1), s1 = sinf(0.5f * d1);
  float c2 = cosf(0.5f * d2), s2 = sinf(0.5f * d2);
  float c3 = cosf(0.5f * d3), s3 = sinf(0.5f * d3);

  v8f ur = {};
  v8f ui = {};
#pragma unroll
  for (int cchunk = 0; cchunk < 4; ++cchunk) {
    int k = cchunk * 4 + 2 * h;  // this lane's K pair within the chunk
    v2f avr = {sVr[col * 16 + k], sVr[col * 16 + k + 1]};
    v2f avi = {sVi[col * 16 + k], sVi[col * 16 + k + 1]};
    v2f bv = {psi_at(k, c0, s0, c1, s1, c2, s2, c3, s3),
              psi_at(k + 1, c0, s0, c1, s1, c2, s2, c3, s3)};
    ur = __builtin_amdgcn_wmma_f32_16x16x4_f32(false, avr, false, bv,
                                               (short)0, ur, false, false);
    ui = __builtin_amdgcn_wmma_f32_16x16x4_f32(false, avi, false, bv,
                                               (short)0, ui, false, false);
  }

  // probabilities + signed Z sums over this lane's 8 amplitude rows
  float z0 = 0.f, z1 = 0.f, z2 = 0.f, z3 = 0.f;
#pragma unroll
  for (int r = 0; r < 8; ++r) {
    int i = r + 8 * h;  // amplitude index (row M)
    float p = ur[r] * ur[r] + ui[r] * ui[r];
    z0 += (i & 8) ? -p : p;
    z1 += (i & 4) ? -p : p;
    z2 += (i & 2) ? -p : p;
    z3 += (i & 1) ? -p : p;
  }
  // combine the two half-wave row groups of this patch column
  z0 += __shfl_xor(z0, 16);
  z1 += __shfl_xor(z1, 16);
  z2 += __shfl_xor(z2, 16);
  z3 += __shfl_xor(z3, 16);

  if (h == 0) {
    float4 zv = {z0, z1, z2, z3};
    *(float4*)(qf + (size_t)patch * 4) = zv;
  }
}

// ---------------------------------------------------------------------------
// Kernel 2: classifier GEMM (2048x784 @ 784x10 + bias) + log_softmax.
// One wave per 16-row tile; N padded 10 -> 16 with zeroed B columns.
// ---------------------------------------------------------------------------
__global__ __launch_bounds__(CWAVES * 32) void classify(
    const float* __restrict__ qf, const float* __restrict__ Wc,
    const float* __restrict__ bc, float* __restrict__ out) {
  __shared__ float lg[CWAVES][16][17];

  int wave = threadIdx.x >> 5;
  int lane = threadIdx.x & 31;
  int h = lane >> 4;
  int col = lane & 15;  // A-row m and B-column o

  int tile = blockIdx.x * CWAVES + wave;
  int row0 = tile * 16;

  int oc = (col < 10) ? col : 9;             // clamp for safe addressing
  float wm = (col < 10) ? 1.0f : 0.0f;       // zero the padded columns
  const float* arow = qf + (size_t)(row0 + col) * 784;
  const float* wrow = Wc + (size_t)oc * 784;

  v8f acc = {};
  for (int kc = 0; kc < 784; kc += 4) {      // uniform trip count (196)
    int k = kc + 2 * h;
    v2f a = {arow[k], arow[k + 1]};
    v2f bb = {wrow[k] * wm, wrow[k + 1] * wm};
    acc = __builtin_amdgcn_wmma_f32_16x16x4_f32(false, a, false, bb,
                                                (short)0, acc, false, false);
  }

#pragma unroll
  for (int r = 0; r < 8; ++r) {
    int m = r + 8 * h;
    lg[wave][m][col] = acc[r] + ((col < 10) ? bc[oc] : 0.0f);
  }
  __syncthreads();

  if (lane < 16) {
    int m = lane;
    const float* L = lg[wave][m];
    float mx = L[0];
#pragma unroll
    for (int o = 1; o < 10; ++o) mx = fmaxf(mx, L[o]);
    float sum = 0.f;
#pragma unroll
    for (int o = 0; o < 10; ++o) sum += expf(L[o] - mx);
    float lse = mx + logf(sum);
    float* op = out + (size_t)(row0 + m) * 10;
#pragma unroll
    for (int o = 0; o < 10; ++o) op[o] = L[o] - lse;
  }
}

// ---------------------------------------------------------------------------
extern "C" void kernel_launch(void* const* d_in, const int* in_sizes, int n_in,
                              void* d_out, int out_size, void* d_ws,
                              size_t ws_size, hipStream_t stream) {
  (void)in_sizes; (void)n_in; (void)out_size; (void)ws_size;
  const float* x  = (const float*)d_in[0];   // (2048,1,28,28)
  const float* w  = (const float*)d_in[1];   // (2,4,2)
  const float* Wc = (const float*)d_in[2];   // (10,784)
  const float* bc = (const float*)d_in[3];   // (10,)
  float* out = (float*)d_out;                // (2048,10)

  float* Vr = (float*)d_ws;                  // 256 floats
  float* Vi = Vr + 256;                      // 256 floats
  float* qf = Vr + 512;                      // 401408*4 floats (~6.4 MB)

  build_unitary<<<1, 16, 0, stream>>>(w, Vr, Vi);
  quanvolve<<<NTILE / QWAVES, QWAVES * 32, 0, stream>>>(x, Vr, qf);
  classify<<<(2048 / 16) / CWAVES, CWAVES * 32, 0, stream>>>(qf, Wc, bc, out);
}